// PairmixLayer_68985764708563
// MI455X (gfx1250) — compile-verified
//
#include <hip/hip_runtime.h>
#include <hip/hip_bf16.h>

// PairmixLayer fused kernel for gfx1250 (MI455X).
//  - radial GEMM mul[n,f,idx] = sum_k W[idx,f,k] * g[n,k]  -> v_wmma_f32_16x16x4_f32
//  - CG tensor product + accumulation                      -> unrolled VALU FMA,
//    CG coefficients read as wave-uniform scalar (SMEM) loads, not LDS.
// One workgroup (256 threads, 8 wave32) handles a 16-n tile and streams all 64 f
// in four chunks of 16, so x/y/out move through HBM exactly once (~1.04 GB total).

typedef float v2f __attribute__((ext_vector_type(2)));
typedef float v8f __attribute__((ext_vector_type(8)));

#define NFEAT 64
#define KBAS  16
#define NMATS 27           // (LX+1)*(LY+1)*(LOUT+1)
#define RCUT  5.0f

__launch_bounds__(256)
__global__ void pairmix_kernel(const float* __restrict__ x,
                               const float* __restrict__ y,
                               const float* __restrict__ r,
                               const float* __restrict__ w,     // [27][64][16]
                               const float* __restrict__ cgc,   // [3][3][3][5][5][5]
                               float* __restrict__ out,
                               int n_total) {
    __shared__ float g_lds[16 * 17];              // padded: stride 17 avoids bank conflicts
    __shared__ float mul_lds[16 * NMATS * 16];    // [n_loc][idx][f_loc]

    const int tid  = threadIdx.x;
    const int lane = tid & 31;
    const int wave = tid >> 5;
    const int n0   = blockIdx.x * 16;

    // ---- Bernstein basis tile g[16 n][16 k] ----
    if (tid < 16) {
        const float binom[16] = {1.f, 15.f, 105.f, 455.f, 1365.f, 3003.f, 5005.f, 6435.f,
                                 6435.f, 5005.f, 3003.f, 1365.f, 455.f, 105.f, 15.f, 1.f};
        int nidx = n0 + tid;
        if (nidx >= n_total) nidx = n_total - 1;
        float u = r[nidx] * (1.0f / RCUT);
        u = fminf(fmaxf(u, 0.0f), 1.0f);
        const float v = 1.0f - u;
        float upow[16];
        float up = 1.0f;
#pragma unroll
        for (int k = 0; k < 16; ++k) { upow[k] = up; up *= u; }
        float vp = 1.0f;
#pragma unroll
        for (int k = 15; k >= 0; --k) { g_lds[tid * 17 + k] = binom[k] * upow[k] * vp; vp *= v; }
    }
    __syncthreads();

    // WMMA lane decomposition (16x16x4 f32, wave32):
    //  A (16x4): lane m=lane&15 holds K = half*2 + {v0,v1}, half = lane>>4
    //  B (4x16): lane col=lane&15 holds K = half*2 + {v0,v1}
    //  C (16x16): VGPR i -> row M = i + 8*half, col N = lane&15
    const int half = lane >> 4;
    const int ncol = lane & 15;

    // A-operand (g tile) is invariant across f-chunks and idx tiles: hoist to regs.
    v2f areg[4];
#pragma unroll
    for (int kc = 0; kc < 4; ++kc) {
        const int kbase = kc * 4 + half * 2;
        areg[kc].x = g_lds[ncol * 17 + kbase];
        areg[kc].y = g_lds[ncol * 17 + kbase + 1];
    }

    // Consumer thread mapping (constant across f-chunks).
    const int n_loc = tid >> 4;
    const int f_loc = tid & 15;
    const int n     = n0 + n_loc;
    const int n_c   = (n < n_total) ? n : (n_total - 1);   // clamped for loads
    const bool valid = (n < n_total);

#pragma unroll 1
    for (int fc = 0; fc < 4; ++fc) {
        // ---- producer: mul tiles via WMMA (waves split the 27 idx tiles) ----
        for (int t = wave; t < NMATS; t += 8) {
            v8f acc = {0.f, 0.f, 0.f, 0.f, 0.f, 0.f, 0.f, 0.f};
#pragma unroll
            for (int kc = 0; kc < 4; ++kc) {
                const int kbase = kc * 4 + half * 2;
                const int f = fc * 16 + ncol;
                v2f b;
                b.x = w[t * (NFEAT * KBAS) + f * KBAS + kbase];
                b.y = w[t * (NFEAT * KBAS) + f * KBAS + kbase + 1];
                acc = __builtin_amdgcn_wmma_f32_16x16x4_f32(
                    false, areg[kc], false, b, (short)0, acc, false, false);
            }
#pragma unroll
            for (int i = 0; i < 8; ++i)
                mul_lds[(i + 8 * half) * (NMATS * 16) + t * 16 + ncol] = acc[i];
        }
        __syncthreads();

        // ---- consumer: one (n,f) pair per thread ----
        {
            const int f = fc * 16 + f_loc;
            const long base = ((long)n_c * NFEAT + f) * 9;
            const float* xp = x + base;
            const float* yp = y + base;
            float xv[9], yv[9], o[9];
#pragma unroll
            for (int a = 0; a < 9; ++a) { xv[a] = xp[a]; yv[a] = yp[a]; o[a] = 0.0f; }

#pragma unroll
            for (int l1 = 0; l1 < 3; ++l1) {
#pragma unroll
                for (int l2 = 0; l2 < 3; ++l2) {
                    const int d1 = 2 * l1 + 1, d2 = 2 * l2 + 1;
                    float p[5][5];
#pragma unroll
                    for (int a = 0; a < d1; ++a)
#pragma unroll
                        for (int b = 0; b < d2; ++b)
                            p[a][b] = xv[l1 * l1 + a] * yv[l2 * l2 + b];
#pragma unroll
                    for (int l3 = 0; l3 < 3; ++l3) {
                        const int d3 = 2 * l3 + 1;
                        const float mval =
                            mul_lds[n_loc * (NMATS * 16) + (l1 * 9 + l2 * 3 + l3) * 16 + f_loc];
                        // wave-uniform constant-indexed reads -> scalar (SMEM) loads
                        const float* cg = cgc + ((l1 * 3 + l2) * 3 + l3) * 125;
#pragma unroll
                        for (int c = 0; c < d3; ++c) {
                            float s = 0.0f;
#pragma unroll
                            for (int a = 0; a < d1; ++a)
#pragma unroll
                                for (int b = 0; b < d2; ++b)
                                    s = fmaf(p[a][b], cg[(a * 5 + b) * 5 + c], s);
                            o[l3 * l3 + c] = fmaf(mval, s, o[l3 * l3 + c]);
                        }
                    }
                }
            }
            if (valid) {
                float* op = out + base;
#pragma unroll
                for (int c = 0; c < 9; ++c) op[c] = o[c];
            }
        }
        __syncthreads();  // before next fc overwrites mul_lds
    }
}

extern "C" void kernel_launch(void* const* d_in, const int* in_sizes, int n_in,
                              void* d_out, int out_size, void* d_ws, size_t ws_size,
                              hipStream_t stream) {
    const float* x   = (const float*)d_in[0];
    const float* y   = (const float*)d_in[1];
    const float* r   = (const float*)d_in[2];
    const float* w   = (const float*)d_in[3];
    const float* cgc = (const float*)d_in[4];
    float* out = (float*)d_out;

    const int N = in_sizes[2];            // r has one element per n
    const int nblk = (N + 15) / 16;
    pairmix_kernel<<<nblk, 256, 0, stream>>>(x, y, r, w, cgc, out, N);
}